// WarpV2_58153857188505
// MI455X (gfx1250) — compile-verified
//
#include <hip/hip_runtime.h>

// WarpV2 / tfa.dense_image_warp on MI455X (gfx1250).
// Bandwidth-bound bilinear gather: ~0.55 GB total HBM traffic -> ~23us floor
// at 23.3 TB/s. WMMA is inapplicable (per-pixel gathered B operands); the
// CDNA5-specific paths used: TDM (tensor_load_to_lds + s_wait_tensorcnt) for
// flow staging, global_prefetch_b8, NT stores to keep the 256 MiB image
// resident in the 192 MB L2, wave32-aware 16-lane-per-pixel coalescing.

typedef float        v4f   __attribute__((ext_vector_type(4)));
typedef unsigned int u32x4 __attribute__((ext_vector_type(4)));
typedef int          i32x4 __attribute__((ext_vector_type(4)));
typedef int          i32x8 __attribute__((ext_vector_type(8)));

#define BB 8
#define HH 256
#define WW 512
#define CC 64
#define THREADS 512
#define PX_PER_BLOCK (THREADS / 16)        // 32 pixels / block
#define FLOW_FLOATS  (2 * PX_PER_BLOCK)    // 64 floats = 256 B tile

__global__ __launch_bounds__(THREADS)
void warpv2_bilinear_kernel(const float* __restrict__ img,
                            const float* __restrict__ flo,
                            float* __restrict__ out) {
    __shared__ float sflow[FLOW_FLOATS];

    const unsigned tid = threadIdx.x;
    const unsigned blk = blockIdx.x;

    // ---- Stage this block's flow tile into LDS via the Tensor Data Mover.
    // Only wave 0 executes the TDM issue (one DMA per issuing wave).
    if (tid < 32) {
        const unsigned long long gaddr =
            (unsigned long long)(const void*)flo +
            (unsigned long long)blk * (FLOW_FLOATS * 4u);

        u32x4 g0;
        g0.x = 1u;                                    // count=1, user mode
        g0.y = 0u;                                    // lds_addr = 0 (sflow base)
        g0.z = (unsigned)(gaddr & 0xFFFFFFFFull);     // global_addr[31:0]
        g0.w = (unsigned)((gaddr >> 32) & 0x1FFFFFFull)
             | 0x80000000u;                           // global_addr[56:32] | type=2

        i32x8 g1;
        g1[0] = 0x00020000;                   // mask=0, data_size=2 (4 B elems)
        g1[1] = (int)(FLOW_FLOATS << 16);     // tensor_dim0.lo
        g1[2] = (int)(1u << 16);              // tensor_dim0.hi=0, tensor_dim1.lo=1
        g1[3] = (int)(FLOW_FLOATS << 16);     // tensor_dim1.hi=0, tile_dim0
        g1[4] = 1;                            // tile_dim1=1, tile_dim2=0
        g1[5] = FLOW_FLOATS;                  // tensor_dim0_stride.lo
        g1[6] = 0;
        g1[7] = 0;

        const i32x4 z4 = {0, 0, 0, 0};
#if __clang_major__ >= 23
        const i32x8 z8 = {0, 0, 0, 0, 0, 0, 0, 0};
        __builtin_amdgcn_tensor_load_to_lds(g0, g1, z4, z4, z8, 0);
#else
        __builtin_amdgcn_tensor_load_to_lds(g0, g1, z4, z4, 0);
#endif
        __builtin_amdgcn_s_wait_tensorcnt(0);

        if (tid == 0) {
            // Speculative prefetch of the next block's flow tile.
            __builtin_prefetch((const char*)flo +
                               (unsigned long long)(blk + 1) *
                                   (FLOW_FLOATS * 4u), 0, 0);
        }
    }
    // The TDM builtin doesn't reference sflow's pointer, so LLVM would prove
    // sflow store-free and fold its loads to undef (observed as ds=0 in the
    // round-1 histogram). Escape the address into an opaque asm that may
    // write memory so the ds_load of the flow tile is real.
    asm volatile("" : : "r"(&sflow[0]) : "memory");
    __syncthreads();

    // ---- Per-thread work: 4 channels (one float4) of one pixel.
    const unsigned pb = tid >> 4;        // pixel within block
    const unsigned c4 = tid & 15;        // float4 channel group, 0..15
    const unsigned p  = blk * PX_PER_BLOCK + pb;   // flat pixel id

    const unsigned x = p & (WW - 1);          // W = 512 = 2^9
    const unsigned y = (p >> 9) & (HH - 1);   // H = 256 = 2^8
    const unsigned b = p >> 17;

    // reference: flow = -flo[...,::-1]; q = grid - flow  =>  q = grid + flo
    const float f0 = sflow[2 * pb + 0];   // flo[...,0] -> adds to x
    const float f1 = sflow[2 * pb + 1];   // flo[...,1] -> adds to y
    const float qx = (float)x + f0;
    const float qy = (float)y + f1;

    float fx = fminf(fmaxf(floorf(qx), 0.0f), (float)(WW - 2));
    float fy = fminf(fmaxf(floorf(qy), 0.0f), (float)(HH - 2));
    const float ax = fminf(fmaxf(qx - fx, 0.0f), 1.0f);
    const float ay = fminf(fmaxf(qy - fy, 0.0f), 1.0f);
    const int ix = (int)fx;
    const int iy = (int)fy;

    const float* p00 =
        img + ((((size_t)b * HH + (size_t)iy) * WW + (size_t)ix) * CC
               + (size_t)c4 * 4);

    const v4f tl = *(const v4f*)(p00);
    const v4f tr = *(const v4f*)(p00 + CC);
    const v4f bl = *(const v4f*)(p00 + (size_t)WW * CC);
    const v4f br = *(const v4f*)(p00 + (size_t)WW * CC + CC);

    const v4f top = tl + ax * (tr - tl);
    const v4f bot = bl + ax * (br - bl);
    const v4f res = top + ay * (bot - top);

    // Write-once output: non-temporal so it doesn't evict img from L2.
    __builtin_nontemporal_store(res,
        (v4f*)(out + (size_t)p * CC + (size_t)c4 * 4));
}

extern "C" void kernel_launch(void* const* d_in, const int* in_sizes, int n_in,
                              void* d_out, int out_size, void* d_ws, size_t ws_size,
                              hipStream_t stream) {
    const float* img = (const float*)d_in[0];   // [8,256,512,64] f32
    const float* flo = (const float*)d_in[1];   // [8,256,512,2]  f32
    float* out = (float*)d_out;                 // [8,256,512,64] f32

    const unsigned total_px = BB * HH * WW;             // 1,048,576
    const unsigned blocks   = total_px / PX_PER_BLOCK;  // 32,768

    warpv2_bilinear_kernel<<<blocks, THREADS, 0, stream>>>(img, flo, out);
}